// LoraLinear_9363028705701
// MI455X (gfx1250) — compile-verified
//
#include <hip/hip_runtime.h>

typedef float v2f __attribute__((ext_vector_type(2)));
typedef float v4f __attribute__((ext_vector_type(4)));
typedef float v8f __attribute__((ext_vector_type(8)));

#define IN_F    1024
#define OUT_F   1024
#define RNK     16
#define BATCH   4096
#define SCALING 2.0f

// ---------------------------------------------------------------------------
// Kernel 1: a[BATCH x 16] = x @ A^T via V_WMMA_F32_16X16X4_F32.
// One wave per 16-batch-row tile; K loop over IN_F in steps of 4.
// A-operand (x):   lane l, vgpr j -> x[b0 + l%16][k0 + 2*(l/16) + j]
// B-operand (A^T): lane l, vgpr j -> A[l%16][k0 + 2*(l/16) + j]
// ---------------------------------------------------------------------------
__global__ __launch_bounds__(32) void lora_compute_a(
    const float* __restrict__ x,
    const float* __restrict__ Alora,
    float* __restrict__ a_out)
{
    const int lane = threadIdx.x;
    const int row  = lane & 15;
    const int half = lane >> 4;
    const int b0   = blockIdx.x * 16;

    const float* xrow = x     + (size_t)(b0 + row) * IN_F + 2 * half;
    const float* arow = Alora + (size_t)row        * IN_F + 2 * half;

    v8f acc = {};
#pragma unroll 8
    for (int k0 = 0; k0 < IN_F; k0 += 4) {
        v2f av = *(const v2f*)(xrow + k0);
        v2f bv = *(const v2f*)(arow + k0);
        acc = __builtin_amdgcn_wmma_f32_16x16x4_f32(
            false, av, false, bv, (short)0, acc, false, false);
    }

    // D layout: vgpr j holds [M = j + 8*half][N = row]
#pragma unroll
    for (int j = 0; j < 8; ++j) {
        const int b = b0 + j + 8 * half;
        a_out[(size_t)b * RNK + row] = acc[j];
    }
}

// ---------------------------------------------------------------------------
// Kernel 2 (fused): 32(batch) x 32(out) tile per wave, 2x2 register-blocked
// f32 WMMA for the base GEMM (each loaded operand feeds 2 WMMAs), then
// bias + LoRA gather epilogue.
// ---------------------------------------------------------------------------
__global__ __launch_bounds__(32) void lora_fused_main(
    const float* __restrict__ x,
    const int*   __restrict__ user_idx,
    const int*   __restrict__ item_idx,
    const float* __restrict__ W_base,
    const float* __restrict__ b_base,
    const float* __restrict__ B_user,
    const float* __restrict__ B_item,
    const float* __restrict__ W_common,
    const float* __restrict__ a_ws,
    float*       __restrict__ out)
{
    const int lane = threadIdx.x;
    const int row  = lane & 15;
    const int half = lane >> 4;
    const int o0   = blockIdx.x * 32;
    const int b0   = blockIdx.y * 32;

    // Resolve gather indices and prefetch the 64B LoRA-B rows now so the
    // HBM-bound gather overlaps the WMMA GEMM loop.
    int uidx[2][8], iidx[2][8];
#pragma unroll
    for (int bb = 0; bb < 2; ++bb) {
#pragma unroll
        for (int j = 0; j < 8; ++j) {
            const int b = b0 + 16 * bb + j + 8 * half;
            uidx[bb][j] = user_idx[b];
            iidx[bb][j] = item_idx[b];
#pragma unroll
            for (int oo = 0; oo < 2; ++oo) {
                const int o_ = o0 + 16 * oo + row;
                __builtin_prefetch(B_user + ((size_t)uidx[bb][j] * OUT_F + o_) * RNK, 0, 0);
                __builtin_prefetch(B_item + ((size_t)iidx[bb][j] * OUT_F + o_) * RNK, 0, 0);
            }
        }
    }

    // Base GEMM: 2x2 accumulators of 16x16; K = 1024 in steps of 4.
    const float* xrow0 = x      + (size_t)(b0 + row)      * IN_F + 2 * half;
    const float* xrow1 = x      + (size_t)(b0 + 16 + row) * IN_F + 2 * half;
    const float* wrow0 = W_base + (size_t)(o0 + row)      * IN_F + 2 * half;
    const float* wrow1 = W_base + (size_t)(o0 + 16 + row) * IN_F + 2 * half;

    v8f acc00 = {}, acc01 = {}, acc10 = {}, acc11 = {};
#pragma unroll 4
    for (int k0 = 0; k0 < IN_F; k0 += 4) {
        v2f a0 = *(const v2f*)(xrow0 + k0);
        v2f a1 = *(const v2f*)(xrow1 + k0);
        v2f w0 = *(const v2f*)(wrow0 + k0);
        v2f w1 = *(const v2f*)(wrow1 + k0);
        acc00 = __builtin_amdgcn_wmma_f32_16x16x4_f32(false, a0, false, w0, (short)0, acc00, false, false);
        acc01 = __builtin_amdgcn_wmma_f32_16x16x4_f32(false, a0, false, w1, (short)0, acc01, false, false);
        acc10 = __builtin_amdgcn_wmma_f32_16x16x4_f32(false, a1, false, w0, (short)0, acc10, false, false);
        acc11 = __builtin_amdgcn_wmma_f32_16x16x4_f32(false, a1, false, w1, (short)0, acc11, false, false);
    }

    // Per-o-column constants: bias and W_common row (o depends on lane%16 + oo).
    float bias[2];
    v4f   wc[2][4];
#pragma unroll
    for (int oo = 0; oo < 2; ++oo) {
        const int o_ = o0 + 16 * oo + row;
        bias[oo] = b_base[o_];
        const float* wcp = W_common + (size_t)o_ * RNK;
#pragma unroll
        for (int q = 0; q < 4; ++q) wc[oo][q] = *(const v4f*)(wcp + 4 * q);
    }

    // Epilogue: for each of the 32 (b,o) pairs this lane owns:
    //   out = base + bias + SCALING * ((Bu + Bi + Wc) . a)
#pragma unroll
    for (int bb = 0; bb < 2; ++bb) {
#pragma unroll
        for (int j = 0; j < 8; ++j) {
            const int b = b0 + 16 * bb + j + 8 * half;

            v4f avq[4];
            const float* ap = a_ws + (size_t)b * RNK;
#pragma unroll
            for (int q = 0; q < 4; ++q) avq[q] = *(const v4f*)(ap + 4 * q);

            const float accv[2] = {
                bb ? ((j < 8) ? ((float*)&acc10)[0] : 0.0f) : 0.0f, 0.0f };
            (void)accv; // (placeholder removed below; direct indexing used)

#pragma unroll
            for (int oo = 0; oo < 2; ++oo) {
                const int o_ = o0 + 16 * oo + row;
                const float* bup = B_user + ((size_t)uidx[bb][j] * OUT_F + o_) * RNK;
                const float* bip = B_item + ((size_t)iidx[bb][j] * OUT_F + o_) * RNK;

                float lora = 0.0f;
#pragma unroll
                for (int q = 0; q < 4; ++q) {
                    v4f buq = *(const v4f*)(bup + 4 * q);
                    v4f biq = *(const v4f*)(bip + 4 * q);
                    v4f wcq = wc[oo][q];
                    v4f aq  = avq[q];
                    lora += (buq.x + biq.x + wcq.x) * aq.x;
                    lora += (buq.y + biq.y + wcq.y) * aq.y;
                    lora += (buq.z + biq.z + wcq.z) * aq.z;
                    lora += (buq.w + biq.w + wcq.w) * aq.w;
                }

                float base;
                if      (bb == 0 && oo == 0) base = acc00[j];
                else if (bb == 0 && oo == 1) base = acc01[j];
                else if (bb == 1 && oo == 0) base = acc10[j];
                else                         base = acc11[j];

                out[(size_t)b * OUT_F + o_] = base + bias[oo] + SCALING * lora;
            }
        }
    }
}

// ---------------------------------------------------------------------------
// Host launcher
// ---------------------------------------------------------------------------
extern "C" void kernel_launch(void* const* d_in, const int* in_sizes, int n_in,
                              void* d_out, int out_size, void* d_ws, size_t ws_size,
                              hipStream_t stream) {
    const float* x   = (const float*)d_in[0];
    const int*   ui  = (const int*)  d_in[1];
    const int*   ii  = (const int*)  d_in[2];
    const float* Wb  = (const float*)d_in[3];
    const float* bb  = (const float*)d_in[4];
    const float* Al  = (const float*)d_in[5];
    const float* Bu  = (const float*)d_in[6];
    const float* Bi  = (const float*)d_in[7];
    const float* Wc  = (const float*)d_in[8];
    float*       out = (float*)d_out;
    float*       aws = (float*)d_ws;   // BATCH*RNK floats = 256 KB scratch

    lora_compute_a<<<BATCH / 16, 32, 0, stream>>>(x, Al, aws);

    dim3 grid(OUT_F / 32, BATCH / 32);
    lora_fused_main<<<grid, 32, 0, stream>>>(x, ui, ii, Wb, bb, Bu, Bi, Wc, aws, out);
}